// InitialUniformAggNodeModel_49976239456343
// MI455X (gfx1250) — compile-verified
//
#include <hip/hip_runtime.h>

// CDNA5 / gfx1250, wave32.
typedef float v2f __attribute__((ext_vector_type(2)));
typedef float v8f __attribute__((ext_vector_type(8)));

// ---------------- zero workspace ----------------
__global__ void zero_kernel(float* __restrict__ p, long long n) {
  long long i = (long long)blockIdx.x * blockDim.x + threadIdx.x;
  long long stride = (long long)gridDim.x * blockDim.x;
  for (; i < n; i += stride) p[i] = 0.0f;
}

// ---------------- scatter-add edges into both endpoints ----------------
// One thread handles a float4 chunk (16 chunks per edge of 64 floats).
// unsafeAtomicAdd -> global_atomic_add_f32 (no-return, STOREcnt); the 25.6MB
// flow buffer is L2-resident, so these are L2 atomics, never HBM RMW.
__global__ __launch_bounds__(256) void scatter_kernel(
    const int* __restrict__ eidx,     // [2, E] int32: row0=past, row1=future
    const float* __restrict__ attr,   // [E, 64]
    float* __restrict__ flow,         // [N, 64]
    int E) {
  long long g = (long long)blockIdx.x * 256 + threadIdx.x;
  int e = (int)(g >> 4);
  int c = (int)(g & 15);
  if (e >= E) return;
  float4 v = ((const float4*)attr)[(long long)e * 16 + c];
  int p = eidx[e];
  int f = eidx[(long long)E + e];
  float* dp = flow + (long long)p * 64 + c * 4;
  float* df = flow + (long long)f * 64 + c * 4;
  unsafeAtomicAdd(dp + 0, v.x);
  unsafeAtomicAdd(dp + 1, v.y);
  unsafeAtomicAdd(dp + 2, v.z);
  unsafeAtomicAdd(dp + 3, v.w);
  unsafeAtomicAdd(df + 0, v.x);
  unsafeAtomicAdd(df + 1, v.y);
  unsafeAtomicAdd(df + 2, v.z);
  unsafeAtomicAdd(df + 3, v.w);
}

// ---------------- fused node MLP via V_WMMA_F32_16X16X4_F32 ----------------
// out = relu(X @ W1 + b1) @ W2 + b2 ; X=[N,64], W1=[64,128], W2=[128,64]
#define W1_PITCH 144   // 128 + 16 -> k and k+2 rows land on disjoint bank groups
#define W2_PITCH 80    // 64 + 16
#define Y_PITCH  132   // 128 + 4  -> per-row bank rotation for A-fragment reads

__global__ __launch_bounds__(128) void mlp_kernel(
    const float* __restrict__ X,
    const float* __restrict__ W1, const float* __restrict__ b1,
    const float* __restrict__ W2, const float* __restrict__ b2,
    float* __restrict__ out, int N) {
  __shared__ float sW1[64 * W1_PITCH];
  __shared__ float sW2[128 * W2_PITCH];
  __shared__ float sB1[128];
  __shared__ float sB2[64];
  __shared__ float sY[4][16 * Y_PITCH];   // per-wave 16x128 intermediate

  const int tid = threadIdx.x;

  // Stage weights/biases once per block.
  for (int i = tid; i < 64 * 128; i += 128) {
    sW1[(i >> 7) * W1_PITCH + (i & 127)] = W1[i];
  }
  for (int i = tid; i < 128 * 64; i += 128) {
    sW2[(i >> 6) * W2_PITCH + (i & 63)] = W2[i];
  }
  if (tid < 128) sB1[tid] = b1[tid];
  if (tid < 64)  sB2[tid] = b2[tid];
  __syncthreads();

  const int wave = tid >> 5;       // 4 waves per block (wave32)
  const int lane = tid & 31;
  const int l16  = lane & 15;
  const int half = lane >> 4;

  const int ntiles = (N + 15) >> 4;
  const int tile = blockIdx.x * 4 + wave;
  if (tile >= ntiles) return;      // wave-uniform: EXEC stays all-1s for WMMA

  // ---- Layer 1: Y[16x128] = relu(Xtile[16x64] @ W1 + b1) ----
  // A-fragment (ISA 32-bit A 16x4): lane L -> M=L%16, VGPR j -> K=k0+2*(L>>4)+j
  int node = tile * 16 + l16;
  int nodeLd = node < N ? node : N - 1;           // clamp (stores are guarded)
  const float* xrow = X + (long long)nodeLd * 64;

  v2f aX[16];
#pragma unroll
  for (int kk = 0; kk < 16; ++kk) {
    aX[kk] = *(const v2f*)(xrow + kk * 4 + 2 * half);   // one b64 load each
  }

  float* yrow = sY[wave];
#pragma unroll
  for (int nt = 0; nt < 8; ++nt) {
    const int ncol = nt * 16 + l16;
    v8f acc = {};
#pragma unroll
    for (int kk = 0; kk < 16; ++kk) {
      const int k0 = kk * 4 + 2 * half;
      v2f bW;                                     // B 4x16: lane L -> N=L%16
      bW.x = sW1[(k0 + 0) * W1_PITCH + ncol];
      bW.y = sW1[(k0 + 1) * W1_PITCH + ncol];
      acc = __builtin_amdgcn_wmma_f32_16x16x4_f32(
          false, aX[kk], false, bW, (short)0, acc, false, false);
    }
    const float bias = sB1[ncol];
#pragma unroll
    for (int r = 0; r < 8; ++r) {                 // D: VGPR r -> row r+8*half
      float y = acc[r] + bias;
      yrow[(r + 8 * half) * Y_PITCH + ncol] = y > 0.0f ? y : 0.0f;
    }
  }

  // ---- Layer 2: out_tile[16x64] = Y @ W2 + b2 ----
  // Re-read Y from LDS in A-fragment layout (contiguous float2 per lane).
  v2f aY[32];
#pragma unroll
  for (int kk = 0; kk < 32; ++kk) {
    aY[kk] = *(const v2f*)(yrow + l16 * Y_PITCH + kk * 4 + 2 * half);
  }
#pragma unroll
  for (int ot = 0; ot < 4; ++ot) {
    const int ncol = ot * 16 + l16;
    v8f acc = {};
#pragma unroll
    for (int kk = 0; kk < 32; ++kk) {
      const int k0 = kk * 4 + 2 * half;
      v2f bW;
      bW.x = sW2[(k0 + 0) * W2_PITCH + ncol];
      bW.y = sW2[(k0 + 1) * W2_PITCH + ncol];
      acc = __builtin_amdgcn_wmma_f32_16x16x4_f32(
          false, aY[kk], false, bW, (short)0, acc, false, false);
    }
    const float bias = sB2[ncol];
#pragma unroll
    for (int r = 0; r < 8; ++r) {
      const int row = tile * 16 + r + 8 * half;
      if (row < N) out[(long long)row * 64 + ncol] = acc[r] + bias;
    }
  }
}

// ---------------- launch ----------------
extern "C" void kernel_launch(void* const* d_in, const int* in_sizes, int n_in,
                              void* d_out, int out_size, void* d_ws, size_t ws_size,
                              hipStream_t stream) {
  // setup_inputs order: edge_index, edge_attr, num_nodes, W1, b1, W2, b2
  const int*   eidx = (const int*)d_in[0];
  const float* attr = (const float*)d_in[1];
  const float* W1   = (const float*)d_in[3];
  const float* b1   = (const float*)d_in[4];
  const float* W2   = (const float*)d_in[5];
  const float* b2   = (const float*)d_in[6];
  float* out  = (float*)d_out;
  float* flow = (float*)d_ws;                 // [N,64] accumulator

  const int E = in_sizes[1] / 64;
  const int N = out_size / 64;                // output is [N,64]

  zero_kernel<<<512, 256, 0, stream>>>(flow, (long long)N * 64);

  const long long work = (long long)E * 16;   // 16 float4-chunks per edge
  const int sblocks = (int)((work + 255) / 256);
  scatter_kernel<<<sblocks, 256, 0, stream>>>(eidx, attr, flow, E);

  const int ntiles = (N + 15) / 16;
  const int mblocks = (ntiles + 3) / 4;       // 4 waves (tiles) per block
  mlp_kernel<<<mblocks, 128, 0, stream>>>(flow, W1, b1, W2, b2, out, N);
}